// ResidualAttentionBlock_4939212391074
// MI455X (gfx1250) — compile-verified
//
#include <hip/hip_runtime.h>
#include <math.h>

// GATv2 + GraphNorm + ELU + residual for MI455X (gfx1250, wave32).
// C = 128 channels, H = 2 heads, HC = 256.

#define CCH 128
#define HCH 256

typedef __attribute__((ext_vector_type(2))) float v2f;
typedef __attribute__((ext_vector_type(8))) float v8f;

__device__ __forceinline__ float lrelu02(float v) { return v > 0.0f ? v : 0.2f * v; }

// ---------------------------------------------------------------------------
// Init: zero accumulators, seg sums, col stats; seg max = -inf.
// ---------------------------------------------------------------------------
__global__ void init_kernel(float* __restrict__ out128, float* __restrict__ segmax,
                            float* __restrict__ segsum, float* __restrict__ colsum,
                            float* __restrict__ varsum, int n_nodes) {
    int idx = blockIdx.x * blockDim.x + threadIdx.x;
    long total = (long)n_nodes * CCH;
    if (idx < total) out128[idx] = 0.0f;
    if (idx < 2 * n_nodes) { segmax[idx] = -__builtin_inff(); segsum[idx] = 0.0f; }
    if (idx < CCH) { colsum[idx] = 0.0f; varsum[idx] = 0.0f; }
}

// ---------------------------------------------------------------------------
// Dual GEMM via V_WMMA_F32_16X16X4_F32:
//   xl = x @ Wl^T + bl, xr = x @ Wr^T + br    ([N,128] x [256,128]^T -> [N,256])
// One wave per 16x16 output tile; A fetched once, reused for both B operands.
// A layout (16x4 f32): lanes 0-15 = M rows, v0/v1 = K0/K1; lanes 16-31 = K2/K3.
// B layout (4x16 f32): lanes 0-15 = N cols, v0/v1 = K0/K1; lanes 16-31 = K2/K3.
// C/D layout: lane&15 = N col; VGPR i = row i (lanes<16) or row i+8 (lanes>=16).
// ---------------------------------------------------------------------------
__global__ void gemm_dual_wmma_kernel(const float* __restrict__ x,
                                      const float* __restrict__ Wl, const float* __restrict__ bl,
                                      const float* __restrict__ Wr, const float* __restrict__ br,
                                      float* __restrict__ xl, float* __restrict__ xr,
                                      int tiles_m) {
    int wave = (blockIdx.x * blockDim.x + threadIdx.x) >> 5;
    int lane = threadIdx.x & 31;
    int total = tiles_m * 16;                  // 16 column tiles of the 256-wide output
    if (wave >= total) return;                 // wave-uniform: EXEC stays all-1 for active waves
    int tm = wave >> 4;
    int tn = wave & 15;
    int m0 = tm << 4, n0 = tn << 4;
    int l15 = lane & 15;
    int ksel = (lane >> 4) << 1;               // 0 for lanes 0-15, 2 for lanes 16-31

    const float* arow  = x  + (size_t)(m0 + l15) * CCH + ksel;
    const float* blrow = Wl + (size_t)(n0 + l15) * CCH + ksel;
    const float* brrow = Wr + (size_t)(n0 + l15) * CCH + ksel;

    v8f accl = {};
    v8f accr = {};
    for (int k = 0; k < CCH; k += 4) {
        v2f a  = *(const v2f*)(arow + k);
        v2f bL = *(const v2f*)(blrow + k);
        v2f bR = *(const v2f*)(brrow + k);
        accl = __builtin_amdgcn_wmma_f32_16x16x4_f32(false, a, false, bL, (short)0, accl, false, false);
        accr = __builtin_amdgcn_wmma_f32_16x16x4_f32(false, a, false, bR, (short)0, accr, false, false);
    }

    int col = n0 + l15;
    float biasl = bl[col];
    float biasr = br[col];
    int mbase = m0 + ((lane >> 4) << 3);
    #pragma unroll
    for (int i = 0; i < 8; ++i) {
        size_t off = (size_t)(mbase + i) * HCH + col;
        xl[off] = accl[i] + biasl;
        xr[off] = accr[i] + biasr;
    }
}

// Scalar tail for rows not covered by full 16-row tiles (unused when N % 16 == 0).
__global__ void gemm_tail_kernel(const float* __restrict__ x,
                                 const float* __restrict__ Wl, const float* __restrict__ bl,
                                 const float* __restrict__ Wr, const float* __restrict__ br,
                                 float* __restrict__ xl, float* __restrict__ xr,
                                 int row_start, int n_nodes) {
    int idx = blockIdx.x * blockDim.x + threadIdx.x;
    int nrows = n_nodes - row_start;
    if (idx >= nrows * HCH) return;
    int row = row_start + idx / HCH;
    int col = idx % HCH;
    const float* xr_ = x + (size_t)row * CCH;
    const float* wl_ = Wl + (size_t)col * CCH;
    const float* wr_ = Wr + (size_t)col * CCH;
    float sl = bl[col], sr = br[col];
    for (int k = 0; k < CCH; ++k) { float xv = xr_[k]; sl += xv * wl_[k]; sr += xv * wr_[k]; }
    xl[(size_t)row * HCH + col] = sl;
    xr[(size_t)row * HCH + col] = sr;
}

// ---------------------------------------------------------------------------
// Edge logits: one wave per edge (incl. self loops).
//   logit[e][h] = att[h] . leaky_relu(xl[src][h] + xr[dst][h])
// Wave-wide shuffle reduction; lane 0 stores + atomic segment max per dst/head.
// ---------------------------------------------------------------------------
__global__ void edge_logits_kernel(const int* __restrict__ ei, int E, int n_nodes,
                                   const float* __restrict__ xl, const float* __restrict__ xr,
                                   const float* __restrict__ att,
                                   float* __restrict__ logits, float* __restrict__ segmax) {
    int wave = (blockIdx.x * blockDim.x + threadIdx.x) >> 5;
    int lane = threadIdx.x & 31;
    int E2 = E + n_nodes;
    if (wave >= E2) return;
    int src, dst;
    if (wave < E) { src = ei[wave]; dst = ei[E + wave]; }
    else { src = wave - E; dst = src; }

    const float4* xls = (const float4*)(xl + (size_t)src * HCH);
    const float4* xrd = (const float4*)(xr + (size_t)dst * HCH);
    const float4* at  = (const float4*)att;

    // head 0: channels lane*4 .. lane*4+3
    float4 a0 = xls[lane],      b0 = xrd[lane],      w0 = at[lane];
    // head 1
    float4 a1 = xls[32 + lane], b1 = xrd[32 + lane], w1 = at[32 + lane];

    float p0 = lrelu02(a0.x + b0.x) * w0.x + lrelu02(a0.y + b0.y) * w0.y +
               lrelu02(a0.z + b0.z) * w0.z + lrelu02(a0.w + b0.w) * w0.w;
    float p1 = lrelu02(a1.x + b1.x) * w1.x + lrelu02(a1.y + b1.y) * w1.y +
               lrelu02(a1.z + b1.z) * w1.z + lrelu02(a1.w + b1.w) * w1.w;

    #pragma unroll
    for (int off = 16; off > 0; off >>= 1) {
        p0 += __shfl_xor(p0, off, 32);
        p1 += __shfl_xor(p1, off, 32);
    }
    if (lane == 0) {
        logits[(size_t)wave * 2 + 0] = p0;
        logits[(size_t)wave * 2 + 1] = p1;
        atomicMax(&segmax[dst * 2 + 0], p0);
        atomicMax(&segmax[dst * 2 + 1], p1);
    }
}

// ---------------------------------------------------------------------------
// exp(logit - segmax[dst]); accumulate segment sums. Thread per edge.
// ---------------------------------------------------------------------------
__global__ void edge_exp_kernel(const int* __restrict__ ei, int E, int n_nodes,
                                float* __restrict__ logits,
                                const float* __restrict__ segmax, float* __restrict__ segsum) {
    int e = blockIdx.x * blockDim.x + threadIdx.x;
    int E2 = E + n_nodes;
    if (e >= E2) return;
    int dst = (e < E) ? ei[E + e] : (e - E);
    float m0 = segmax[dst * 2 + 0];
    float m1 = segmax[dst * 2 + 1];
    float q0 = __expf(logits[(size_t)e * 2 + 0] - m0);
    float q1 = __expf(logits[(size_t)e * 2 + 1] - m1);
    logits[(size_t)e * 2 + 0] = q0;
    logits[(size_t)e * 2 + 1] = q1;
    atomicAdd(&segsum[dst * 2 + 0], q0);
    atomicAdd(&segsum[dst * 2 + 1], q1);
}

// ---------------------------------------------------------------------------
// Weighted scatter-add with head mean folded in:
//   out128[dst][c] += 0.5 * (alpha0*xl[src][0][c] + alpha1*xl[src][1][c])
// One wave per edge, float4 per lane, f32 L2 atomics.
// ---------------------------------------------------------------------------
__global__ void edge_scatter_kernel(const int* __restrict__ ei, int E, int n_nodes,
                                    const float* __restrict__ logits, const float* __restrict__ segsum,
                                    const float* __restrict__ xl, float* __restrict__ out128) {
    int wave = (blockIdx.x * blockDim.x + threadIdx.x) >> 5;
    int lane = threadIdx.x & 31;
    int E2 = E + n_nodes;
    if (wave >= E2) return;
    int src, dst;
    if (wave < E) { src = ei[wave]; dst = ei[E + wave]; }
    else { src = wave - E; dst = src; }

    __builtin_prefetch(xl + (size_t)src * HCH, 0, 3);   // global_prefetch_b8

    float a0 = logits[(size_t)wave * 2 + 0] / segsum[dst * 2 + 0];
    float a1 = logits[(size_t)wave * 2 + 1] / segsum[dst * 2 + 1];

    const float4* xs = (const float4*)(xl + (size_t)src * HCH);
    float4 u = xs[lane];        // head 0
    float4 v = xs[32 + lane];   // head 1
    float4 r;
    r.x = 0.5f * (a0 * u.x + a1 * v.x);
    r.y = 0.5f * (a0 * u.y + a1 * v.y);
    r.z = 0.5f * (a0 * u.z + a1 * v.z);
    r.w = 0.5f * (a0 * u.w + a1 * v.w);

    float* o = out128 + (size_t)dst * CCH + (lane << 2);
    atomicAdd(o + 0, r.x);
    atomicAdd(o + 1, r.y);
    atomicAdd(o + 2, r.z);
    atomicAdd(o + 3, r.w);
}

// ---------------------------------------------------------------------------
// GraphNorm pass 1: add conv bias, accumulate column sums.
// ---------------------------------------------------------------------------
__global__ void node_bias_colsum_kernel(float* __restrict__ out128, const float* __restrict__ bias,
                                        float* __restrict__ colsum, int n_nodes) {
    int idx = blockIdx.x * blockDim.x + threadIdx.x;
    if (idx >= n_nodes * CCH) return;
    int c = idx & (CCH - 1);
    float v = out128[idx] + bias[c];
    out128[idx] = v;
    atomicAdd(&colsum[c], v);
}

// GraphNorm pass 2: center by mean_scale*mu, accumulate variance sums.
__global__ void node_center_kernel(float* __restrict__ out128, const float* __restrict__ colsum,
                                   const float* __restrict__ mean_scale, float* __restrict__ varsum,
                                   int n_nodes) {
    int idx = blockIdx.x * blockDim.x + threadIdx.x;
    if (idx >= n_nodes * CCH) return;
    int c = idx & (CCH - 1);
    float mu = colsum[c] * (1.0f / (float)n_nodes);
    float o = out128[idx] - mean_scale[c] * mu;
    out128[idx] = o;
    atomicAdd(&varsum[c], o * o);
}

// GraphNorm pass 3: scale/shift, ELU, residual -> d_out.
__global__ void node_final_kernel(const float* __restrict__ out128, const float* __restrict__ varsum,
                                  const float* __restrict__ gw, const float* __restrict__ gb,
                                  const float* __restrict__ x, float* __restrict__ out, int n_nodes) {
    int idx = blockIdx.x * blockDim.x + threadIdx.x;
    if (idx >= n_nodes * CCH) return;
    int c = idx & (CCH - 1);
    float var = varsum[c] * (1.0f / (float)n_nodes);
    float rstd = rsqrtf(var + 1e-5f);
    float o = gw[c] * out128[idx] * rstd + gb[c];
    o = o > 0.0f ? o : (__expf(o) - 1.0f);     // ELU(alpha=1)
    out[idx] = o + x[idx];                     // residual
}

// ---------------------------------------------------------------------------
// Host launcher
// ---------------------------------------------------------------------------
extern "C" void kernel_launch(void* const* d_in, const int* in_sizes, int n_in,
                              void* d_out, int out_size, void* d_ws, size_t ws_size,
                              hipStream_t stream) {
    const float* x    = (const float*)d_in[0];
    const int*   ei   = (const int*)d_in[1];
    const float* Wl   = (const float*)d_in[2];
    const float* bl   = (const float*)d_in[3];
    const float* Wr   = (const float*)d_in[4];
    const float* br   = (const float*)d_in[5];
    const float* att  = (const float*)d_in[6];
    const float* bias = (const float*)d_in[7];
    const float* gw   = (const float*)d_in[8];
    const float* gb   = (const float*)d_in[9];
    const float* gms  = (const float*)d_in[10];
    float* out = (float*)d_out;

    const int n_nodes = in_sizes[0] / CCH;
    const int E  = in_sizes[1] / 2;
    const int E2 = E + n_nodes;

    // Carve workspace
    uint8_t* ws = (uint8_t*)d_ws;
    size_t off = 0;
    auto alloc = [&](size_t bytes) -> float* {
        float* p = (float*)(ws + off);
        off += (bytes + 255) & ~(size_t)255;
        return p;
    };
    float* xl     = alloc((size_t)n_nodes * HCH * sizeof(float));
    float* xr     = alloc((size_t)n_nodes * HCH * sizeof(float));
    float* logits = alloc((size_t)E2 * 2 * sizeof(float));
    float* segmax = alloc((size_t)n_nodes * 2 * sizeof(float));
    float* segsum = alloc((size_t)n_nodes * 2 * sizeof(float));
    float* out128 = alloc((size_t)n_nodes * CCH * sizeof(float));
    float* colsum = alloc(CCH * sizeof(float));
    float* varsum = alloc(CCH * sizeof(float));
    (void)ws_size; (void)n_in; (void)out_size;

    const int BLK = 256;
    long node_elems = (long)n_nodes * CCH;

    // 0) init accumulators
    init_kernel<<<(int)((node_elems + BLK - 1) / BLK), BLK, 0, stream>>>(
        out128, segmax, segsum, colsum, varsum, n_nodes);

    // 1) dual GEMM (WMMA): xl / xr
    int tiles_m = n_nodes >> 4;
    if (tiles_m > 0) {
        int waves = tiles_m * 16;
        int blocks = (waves + 7) / 8;          // 8 waves per 256-thread block
        gemm_dual_wmma_kernel<<<blocks, BLK, 0, stream>>>(x, Wl, bl, Wr, br, xl, xr, tiles_m);
    }
    int row_start = tiles_m << 4;
    if (row_start < n_nodes) {
        int tail_elems = (n_nodes - row_start) * HCH;
        gemm_tail_kernel<<<(tail_elems + BLK - 1) / BLK, BLK, 0, stream>>>(
            x, Wl, bl, Wr, br, xl, xr, row_start, n_nodes);
    }

    // 2) edge logits + segment max
    int eblocks_wave = (E2 + 7) / 8;           // one wave per edge
    edge_logits_kernel<<<eblocks_wave, BLK, 0, stream>>>(ei, E, n_nodes, xl, xr, att, logits, segmax);

    // 3) exp + segment sum
    edge_exp_kernel<<<(E2 + BLK - 1) / BLK, BLK, 0, stream>>>(ei, E, n_nodes, logits, segmax, segsum);

    // 4) weighted scatter-add (head mean folded in)
    edge_scatter_kernel<<<eblocks_wave, BLK, 0, stream>>>(ei, E, n_nodes, logits, segsum, xl, out128);

    // 5) GraphNorm + ELU + residual
    int nblocks = (int)((node_elems + BLK - 1) / BLK);
    node_bias_colsum_kernel<<<nblocks, BLK, 0, stream>>>(out128, bias, colsum, n_nodes);
    node_center_kernel<<<nblocks, BLK, 0, stream>>>(out128, colsum, gms, varsum, n_nodes);
    node_final_kernel<<<nblocks, BLK, 0, stream>>>(out128, varsum, gw, gb, x, out, n_nodes);
}